// CIFSampler_78357383348616
// MI455X (gfx1250) — compile-verified
//
#include <hip/hip_runtime.h>
#include <stdint.h>

// CIF sampler for MI455X (gfx1250).
// Pure bandwidth op (~170 MB -> ~7-10us @ 23.3 TB/s). No matmul FLOPs => no WMMA.
// CDNA5 paths used:
//   - TDM gather-mode descriptors (tensor_load_to_lds / tensor_store_from_lds, TENSORcnt)
//   - async LDS->global stores (global_store_async_from_lds_b128, ASYNCcnt)
//   - split wait counters (s_wait_tensorcnt / s_wait_asynccnt / s_wait_dscnt)

#define B_FIXED 16
#define DDIM    512          // feature dim (floats); row = 2 KB
#define GROUP   16           // output rows per block == rows per TDM gather descriptor
#define THRESH  1.0f

typedef unsigned int u32x4 __attribute__((ext_vector_type(4)));
typedef unsigned int u32x8 __attribute__((ext_vector_type(8)));

// ---- CDNA5 primitives (inline asm; portable across ROCm7.2 / amdgpu-toolchain) ----
__device__ __forceinline__ void tdm_load_gather(u32x4 g0, u32x8 g1, u32x4 g2, u32x4 g3) {
  asm volatile("tensor_load_to_lds %0, %1, %2, %3"
               :: "s"(g0), "s"(g1), "s"(g2), "s"(g3) : "memory");
}
__device__ __forceinline__ void tdm_store(u32x4 g0, u32x8 g1) {
  asm volatile("tensor_store_from_lds %0, %1" :: "s"(g0), "s"(g1) : "memory");
}
__device__ __forceinline__ void async_store_b128(void* gaddr, unsigned lds_off) {
  asm volatile("global_store_async_from_lds_b128 %0, %1, off"
               :: "v"(gaddr), "v"(lds_off) : "memory");
}
__device__ __forceinline__ void wait_tensor0() { asm volatile("s_wait_tensorcnt 0" ::: "memory"); }
__device__ __forceinline__ void wait_async0()  { asm volatile("s_wait_asynccnt 0" ::: "memory"); }
__device__ __forceinline__ void wait_ds0()     { asm volatile("s_wait_dscnt 0"    ::: "memory"); }

__device__ __forceinline__ unsigned ballot32(bool p) {
#if __has_builtin(__builtin_amdgcn_ballot_w32)
  return __builtin_amdgcn_ballot_w32(p);
#else
  return (unsigned)__ballot(p);
#endif
}

// ---------------------------------------------------------------------------
// Gather kernel: 1 wave / block, GROUP=16 output rows / block, 32 KB LDS stage.
// grid = (T/GROUP, B): b = blockIdx.y, j0 = blockIdx.x*GROUP  (no int division).
// Valid rows  : TDM gather-mode load (row indices idx[j0..j0+v)) -> LDS,
//               then TDM 2D store LDS -> out rows [j0, j0+v).
// Zero rows   : zero 2 KB of LDS once (ds), then async-store it over each row.
// All descriptor fields are wave-uniform (blockIdx + scalar loads) -> SALU.
// ---------------------------------------------------------------------------
__global__ void __launch_bounds__(32)
cif_gather_tdm_kernel(const float* __restrict__ enc,
                      float* __restrict__ out,
                      const int* __restrict__ ws_nf,
                      const int* __restrict__ ws_idx,
                      int T) {
  __shared__ __align__(16) float tile[GROUP * DDIM];   // 32 KB staging

  const int lane = threadIdx.x;                        // 0..31
  const int b  = blockIdx.y;                           // batch
  const int j0 = blockIdx.x * GROUP;                   // first output row here

  const int nf = *ws_nf;                               // uniform scalar load
  int v;                                               // # valid (gathered) rows here
  if (nf == 0) {
    v = (j0 == 0) ? 1 : 0;                             // fallback: emit enc[:,0,:] at j==0
  } else {
    v = nf - j0;
    v = v < 0 ? 0 : (v > GROUP ? GROUP : v);
  }

  const unsigned lds_base = (unsigned)(uintptr_t)&tile[0];     // low 32b = LDS byte offset
  const uint64_t src_base = (uint64_t)(uintptr_t)(enc + (size_t)b * T * DDIM);
  const uint64_t dst_base = (uint64_t)(uintptr_t)(out + ((size_t)b * T + j0) * DDIM);

  if (v > 0) {
    // ---- pack 16-bit gather row indices (unused slots -> 0) ----
    unsigned pk[8];
#pragma unroll
    for (int q = 0; q < 8; ++q) {
      const int i0 = 2 * q, i1 = 2 * q + 1;
      const unsigned lo = (nf > 0 && i0 < v) ? (unsigned)ws_idx[j0 + i0] : 0u;
      const unsigned hi = (nf > 0 && i1 < v) ? (unsigned)ws_idx[j0 + i1] : 0u;
      pk[q] = (lo & 0xFFFFu) | (hi << 16);
    }

    // ---- D# group1: shared geometry for load & store ----
    // s0: data_size=4B (code 2) @bits17:16 ; workgroup_mask=0 (not in cluster)
    // s1: tensor_dim0[15:0]=512 @bits63:48
    // s2: tensor_dim1[15:0]=T   @bits95:80
    // s3: tile_dim0=512         @bits127:112
    // s4: tile_dim1=v (gather: #valid indices / store: tile rows)
    // s5: tensor_dim0_stride=512
    const u32x8 g1 = { 0x00020000u,
                       (unsigned)DDIM << 16,
                       (unsigned)T << 16,
                       (unsigned)DDIM << 16,
                       (unsigned)v,
                       (unsigned)DDIM,
                       0u, 0u };

    // ---- load descriptor: gather mode, 16-bit indices ----
    const u32x4 g0l = { 0x80000001u,                       // count=1 | gather_mode
                        lds_base,
                        (unsigned)src_base,
                        (unsigned)(src_base >> 32) | 0x80000000u };  // type=2 @bits127:126
    const u32x4 g2 = { pk[0], pk[1], pk[2], pk[3] };
    const u32x4 g3 = { pk[4], pk[5], pk[6], pk[7] };

    tdm_load_gather(g0l, g1, g2, g3);                      // HBM -(DMA gather)-> LDS
    wait_tensor0();                                        // rows landed in LDS

    // ---- store descriptor: plain 2D tile, contiguous dst rows ----
    const u32x4 g0s = { 0x00000001u,                       // count=1
                        lds_base,
                        (unsigned)dst_base,
                        (unsigned)(dst_base >> 32) | 0x80000000u };
    tdm_store(g0s, g1);                                    // LDS -(DMA)-> HBM
    wait_tensor0();                                        // LDS reusable below
  }

  if (v < GROUP) {
    // ---- zero remaining rows: zero 2 KB of LDS once, async-store it per row ----
    const float4 zz = make_float4(0.f, 0.f, 0.f, 0.f);
#pragma unroll
    for (int q = 0; q < 4; ++q)
      *(float4*)&tile[(lane + q * 32) * 4] = zz;           // ds_store_b128
    wait_ds0();

    for (int r = v; r < GROUP; ++r) {
      float* rowp = out + ((size_t)b * T + j0 + r) * DDIM;
#pragma unroll
      for (int q = 0; q < 4; ++q) {
        const int slot = lane + q * 32;                    // 128 float4 slots / row
        async_store_b128(rowp + slot * 4,
                         (unsigned)(uintptr_t)&tile[slot * 4]);
      }
    }
    wait_async0();                                         // drain (endpgm also waits idle)
  }
}

// ---------------------------------------------------------------------------
// Scan kernel: single wave32, lane b handles batch b.
// Serial integrate-and-fire recurrence (inherent to the algorithm); the only
// loop-carried chain is add/cmp/select + ballot; alpha loads are independent.
// ---------------------------------------------------------------------------
__global__ void cif_scan_kernel(const float* __restrict__ alphas,
                                const int* __restrict__ tgt_len,
                                int T,
                                int* __restrict__ ws_nf,
                                int* __restrict__ ws_idx,
                                float* __restrict__ out_tail) {
  const int lane = threadIdx.x;            // blockDim.x == 32 (one wave)
  const bool active = lane < B_FIXED;
  const float* ap = alphas + (size_t)(active ? lane : 0) * (size_t)T;

  float sum = 0.f;
#pragma unroll 8
  for (int t = 0; t < T; ++t) sum += ap[t];

  float scale = (float)(*tgt_len) / (sum + 1e-8f);
  if (!active) scale = 0.f;

  float integ = 0.f;
  int cnt = 0;
  for (int t = 0; t < T; ++t) {
    float a = ap[t] * scale;               // load is chain-independent
    integ += a;
    bool fire = integ >= THRESH;
    integ = fire ? (integ - THRESH) : integ;
    unsigned m = ballot32(fire);           // ANY over the 16 batch lanes
    if (m != 0u) {
      if (lane == 0) ws_idx[cnt] = t;
      ++cnt;                               // uniform across the wave
    }
  }
  if (lane == 0) {
    *ws_nf = cnt;
    *out_tail = (float)cnt;                // second output (num_fires)
  }
}

// ---------------------------------------------------------------------------
extern "C" void kernel_launch(void* const* d_in, const int* in_sizes, int n_in,
                              void* d_out, int out_size, void* d_ws, size_t ws_size,
                              hipStream_t stream) {
  (void)n_in; (void)out_size; (void)ws_size;
  const float* enc    = (const float*)d_in[0];   // [B,T,D] f32
  const float* alphas = (const float*)d_in[1];   // [B,T]   f32
  const int*   tgt    = (const int*)d_in[2];     // scalar int

  float* out = (float*)d_out;                    // [B,T,D] flat + 1 tail slot (num_fires)

  const int B = B_FIXED;
  const int T = in_sizes[1] / B;                 // 4096
  (void)(in_sizes[0] / in_sizes[1]);             // D == DDIM == 512 per reference

  int* ws_nf  = (int*)d_ws;                      // [0]: num_fires
  int* ws_idx = ws_nf + 4;                       // +16 B: idx[T]

  cif_scan_kernel<<<1, 32, 0, stream>>>(alphas, tgt, T, ws_nf, ws_idx,
                                        out + (size_t)in_sizes[0]);

  dim3 grid(T / GROUP, B);                       // 256 x 16 blocks, 1 wave each
  cif_gather_tdm_kernel<<<grid, 32, 0, stream>>>(enc, out, ws_nf, ws_idx, T);
}